// LearnableAlphaWithEpsilon_48936857371170
// MI455X (gfx1250) — compile-verified
//
#include <hip/hip_runtime.h>
#include <stdint.h>

// LearnableAlphaWithEpsilon: u = drelu(x, a) * (1 - a); out = x * prod of u over
// 5-point cross stencil with 1.0 padding. Memory-bound (AI ~0.13 FLOP/B):
// optimize the data path, not math. CDNA5 path: async DMA global->LDS
// (ASYNCcnt), LDS-tiled stencil (5x reuse of u), b128 coalesced stores.

#define EPSILON 0.01f

namespace {
constexpr int Cn = 128;           // channels
constexpr int Hn = 64;            // height
constexpr int Wn = 64;            // width
constexpr int TH = 16;            // tile height (rows per tile)
constexpr int NTILE = Hn / TH;    // 4 tiles per plane
constexpr int HROWS = TH + 2;     // 18 rows incl. top/bottom halo
constexpr int TASKS = HROWS * (Wn / 4); // 288 float4 chunks per tile
constexpr int BLOCK = 256;        // 8 wave32
}

__device__ __forceinline__ float u_of(float x, float a) {
  // gate = (a > EPS); drelu = gate ? (x > 0) : 1 ; u = drelu * (1 - a)
  float drelu = (a > EPSILON) ? ((x > 0.0f) ? 1.0f : 0.0f) : 1.0f;
  return drelu * (1.0f - a);
}

__global__ __launch_bounds__(BLOCK)
void learnable_alpha_stencil_kernel(const float* __restrict__ x,
                                    const float* __restrict__ alphas,
                                    float* __restrict__ out) {
  // Staging buffers (double-buffered) for x and alpha, plus the u tile with
  // constant-1.0 column pads at [*][0] and [*][Wn+1].
  __shared__ __align__(16) float ls_x[2][HROWS][Wn];
  __shared__ __align__(16) float ls_a[2][HROWS][Wn];
  __shared__ __align__(16) float ls_u[HROWS][Wn + 2];

  const int tid = threadIdx.x;
  const int plane = blockIdx.x;            // b*C + c
  const int aplane = plane & (Cn - 1);     // c  (alpha broadcast over batch)
  const long planeBase = (long)plane * (Hn * Wn);
  const long alphaBase = (long)aplane * (Hn * Wn);

  // Constant column pads for u (written once; u-compute never touches them).
  if (tid < HROWS) {
    ls_u[tid][0] = 1.0f;
    ls_u[tid][Wn + 1] = 1.0f;
  }

  // Issue async DMA (global -> LDS, no VGPR round trip) for one tile's x and
  // alpha rows, including the vertical halo. Out-of-plane rows are skipped
  // (u for those rows is written as 1.0 later).
  auto stage = [&](int tile, int buf) {
    const int r0 = tile * TH;
    for (int idx = tid; idx < TASKS; idx += BLOCK) {
      const int row = idx >> 4;            // 0..17
      const int c0 = (idx & 15) << 2;      // 0,4,...,60
      const int gr = r0 - 1 + row;         // plane row
      if (gr >= 0 && gr < Hn) {
        const float* gx = x + planeBase + (long)gr * Wn + c0;
        const float* ga = alphas + alphaBase + (long)gr * Wn + c0;
        uint32_t lx = (uint32_t)(uintptr_t)(&ls_x[buf][row][c0]);
        uint32_t la = (uint32_t)(uintptr_t)(&ls_a[buf][row][c0]);
        uint64_t ax = (uint64_t)(uintptr_t)gx;
        uint64_t aa = (uint64_t)(uintptr_t)ga;
        asm volatile("global_load_async_to_lds_b128 %0, %1, off"
                     :: "v"(lx), "v"(ax) : "memory");
        asm volatile("global_load_async_to_lds_b128 %0, %1, off"
                     :: "v"(la), "v"(aa) : "memory");
      }
    }
  };

  // Prologue: kick off tile 0 DMA.
  stage(0, 0);

  for (int t = 0; t < NTILE; ++t) {
    const int cur = t & 1;
    const int r0 = t * TH;

    // Wait for this wave's outstanding async DMA, then block-barrier so every
    // wave sees the staged tile.
    asm volatile("s_wait_asynccnt 0x0" ::: "memory");
    __syncthreads();

    // Prefetch next tile into the other buffer; overlaps with compute below.
    if (t + 1 < NTILE) stage(t + 1, cur ^ 1);

    // Compute u for the 18-row halo tile into LDS (1.0 for out-of-plane rows).
    for (int idx = tid; idx < TASKS; idx += BLOCK) {
      const int row = idx >> 4;
      const int c0 = (idx & 15) << 2;
      const int gr = r0 - 1 + row;
      if (gr >= 0 && gr < Hn) {
        const float4 xv = *reinterpret_cast<const float4*>(&ls_x[cur][row][c0]);
        const float4 av = *reinterpret_cast<const float4*>(&ls_a[cur][row][c0]);
        ls_u[row][c0 + 1] = u_of(xv.x, av.x);
        ls_u[row][c0 + 2] = u_of(xv.y, av.y);
        ls_u[row][c0 + 3] = u_of(xv.z, av.z);
        ls_u[row][c0 + 4] = u_of(xv.w, av.w);
      } else {
        ls_u[row][c0 + 1] = 1.0f;
        ls_u[row][c0 + 2] = 1.0f;
        ls_u[row][c0 + 3] = 1.0f;
        ls_u[row][c0 + 4] = 1.0f;
      }
    }
    __syncthreads();

    // Stencil: each thread emits one aligned float4 of output.
    {
      const int row = tid >> 4;            // 0..15 within tile
      const int c0 = (tid & 15) << 2;      // output column base
      const int ur = row + 1;              // u-tile row for this output row
      const float4 xc =
          *reinterpret_cast<const float4*>(&ls_x[cur][row + 1][c0]);
      float p[4];
#pragma unroll
      for (int j = 0; j < 4; ++j) {
        const int c = c0 + j;              // ls_u col index of center is c+1
        p[j] = ls_u[ur][c] * ls_u[ur][c + 1] * ls_u[ur][c + 2] *
               ls_u[ur - 1][c + 1] * ls_u[ur + 1][c + 1];
      }
      float4 o;
      o.x = p[0] * xc.x;
      o.y = p[1] * xc.y;
      o.z = p[2] * xc.z;
      o.w = p[3] * xc.w;
      *reinterpret_cast<float4*>(out + planeBase + (long)(r0 + row) * Wn + c0) = o;
    }
    // Protect ls_u and ls_x[cur] from being overwritten before all waves are
    // done reading them (next iteration's DMA / u-compute).
    __syncthreads();
  }
}

extern "C" void kernel_launch(void* const* d_in, const int* in_sizes, int n_in,
                              void* d_out, int out_size, void* d_ws, size_t ws_size,
                              hipStream_t stream) {
  (void)in_sizes; (void)n_in; (void)d_ws; (void)ws_size; (void)out_size;
  const float* x = (const float*)d_in[0];       // (32,128,64,64) f32
  const float* alphas = (const float*)d_in[1];  // (1,128,64,64) f32
  float* out = (float*)d_out;                   // (32,128,64,64) f32

  const int planes = 32 * Cn;                   // B*C = 4096 blocks
  learnable_alpha_stencil_kernel<<<planes, BLOCK, 0, stream>>>(x, alphas, out);
}